// LINE_46462956208853
// MI455X (gfx1250) — compile-verified
//
#include <hip/hip_runtime.h>
#include <math.h>

#define DIM      128
#define NSTEPS   100
#define BATCH    1024
#define KTGT     6
#define ALPHA    0.025f
#define VSTRIDE  130   // padded LDS row stride (floats): banks (2e+c)%64 distinct per edge row

typedef __attribute__((ext_vector_type(2))) float v2f;
typedef __attribute__((ext_vector_type(8))) float v8f;

// ---------------------------------------------------------------------------
// Kernel A: per block = 16 edges. Stage U(16x128) + V(6x16x128) in LDS,
// compute dots with V_WMMA_F32_16X16X4_F32 (diag of 16x16 product), then
// vec_error = sum_k g_k * v_k written to workspace.
// ---------------------------------------------------------------------------
__global__ __launch_bounds__(256) void line_compute_err(
    const float* __restrict__ emb,
    const int*   __restrict__ u_idx,    // [BATCH] for this step
    const int*   __restrict__ tgt_idx,  // [BATCH*KTGT] for this step
    float*       __restrict__ err)      // [BATCH*DIM]
{
    __shared__ float Ulds[16 * VSTRIDE];
    __shared__ float Vlds[KTGT * 16 * VSTRIDE];
    __shared__ float Glds[16 * KTGT];

    const int tid = threadIdx.x;
    const int b0  = blockIdx.x * 16;

    // ---- stage U rows (coalesced b128 global loads, b64 LDS stores) ----
    for (int q = tid; q < 16 * 32; q += 256) {
        const int e  = q >> 5;
        const int c4 = q & 31;
        const int row = u_idx[b0 + e];
        const float4 v = ((const float4*)(emb + (size_t)row * DIM))[c4];
        float* dst = &Ulds[e * VSTRIDE + c4 * 4];
        ((float2*)dst)[0] = make_float2(v.x, v.y);
        ((float2*)dst)[1] = make_float2(v.z, v.w);
    }
    // ---- stage V rows: layout Vlds[k][e][d] ----
    for (int q = tid; q < KTGT * 16 * 32; q += 256) {
        const int r  = q >> 5;
        const int c4 = q & 31;
        const int k  = r >> 4;
        const int e  = r & 15;
        const int row = tgt_idx[(b0 + e) * KTGT + k];
        const float4 v = ((const float4*)(emb + (size_t)row * DIM))[c4];
        float* dst = &Vlds[(k * 16 + e) * VSTRIDE + c4 * 4];
        ((float2*)dst)[0] = make_float2(v.x, v.y);
        ((float2*)dst)[1] = make_float2(v.z, v.w);
    }
    __syncthreads();

    const int wid  = tid >> 5;
    const int lane = tid & 31;

    if (wid < KTGT) {                 // waves 0..5: one k each (wave-uniform)
        const int k = wid;
        v8f c = {};
        // A operand (16x4 f32): lane<16 holds {K0,K1} of row=lane,
        //                       lane>=16 holds {K2,K3} of row=lane-16.
        const int eA = lane & 15;
        const int aK = (lane >> 4) * 2;
        // B operand (4x16 f32): vgpr0 = rows K0(l0-15)/K1(l16-31),
        //                       vgpr1 = rows K2/K3.  B[j][n] = V[n][k][d0+j].
        const int jB = lane >> 4;
        const float* urow = &Ulds[eA * VSTRIDE];
        const float* vrow = &Vlds[(k * 16 + eA) * VSTRIDE];
        #pragma unroll
        for (int d0 = 0; d0 < DIM; d0 += 4) {
            v2f a; a.x = urow[d0 + aK];     a.y = urow[d0 + aK + 1];
            v2f b; b.x = vrow[d0 + jB];     b.y = vrow[d0 + jB + 2];
            c = __builtin_amdgcn_wmma_f32_16x16x4_f32(
                    false, a, false, b, (short)0, c, false, false);
        }
        // diag(C): c[r]@lane r -> m=r ; c[r]@lane r+24 -> m=r+8
        #pragma unroll
        for (int r = 0; r < 8; ++r) {
            const int m = (lane == r) ? r : ((lane == r + 24) ? (r + 8) : -1);
            if (m >= 0) {
                const float dot = c[r];
                const float f   = 1.0f / (1.0f + __expf(-dot));
                const float lbl = (k == 0) ? 1.0f : 0.0f;
                Glds[m * KTGT + k] = ALPHA * (lbl - f);
            }
        }
    }
    __syncthreads();

    // ---- vec_error: thread -> edge e=tid>>4, 8-float segment seg=tid&15 ----
    {
        const int e   = tid >> 4;
        const int seg = tid & 15;
        float acc[8];
        #pragma unroll
        for (int j = 0; j < 8; ++j) acc[j] = 0.0f;
        #pragma unroll
        for (int k = 0; k < KTGT; ++k) {
            const float gk = Glds[e * KTGT + k];
            const float* vr = &Vlds[(k * 16 + e) * VSTRIDE + seg * 8];
            #pragma unroll
            for (int j = 0; j < 8; ++j) acc[j] += gk * vr[j];
        }
        float* out = err + (size_t)(b0 + e) * DIM + seg * 8;
        #pragma unroll
        for (int j = 0; j < 8; ++j) out[j] = acc[j];
    }
}

// ---------------------------------------------------------------------------
// Kernel B: scatter  emb[u[b]] += err[b]  with hardware f32 global atomics
// (u may contain duplicates within a batch).
// ---------------------------------------------------------------------------
__global__ __launch_bounds__(256) void line_scatter(
    float* __restrict__ emb,
    const int* __restrict__ u_idx,
    const float* __restrict__ err)
{
    const int idx = blockIdx.x * 256 + threadIdx.x;   // [0, BATCH*DIM)
    const int b = idx >> 7;
    const int d = idx & (DIM - 1);
    const float v = err[idx];
    const int row = u_idx[b];
    unsafeAtomicAdd(&emb[(size_t)row * DIM + d], v);  // -> global_atomic_add_f32
}

// ---------------------------------------------------------------------------
// Kernel C: L2 row-normalize, one wave32 per row, float4 traffic, shfl reduce.
// ---------------------------------------------------------------------------
__global__ __launch_bounds__(256) void line_normalize(
    float* __restrict__ emb, int nrows)
{
    const int wid  = threadIdx.x >> 5;
    const int lane = threadIdx.x & 31;
    const int row  = blockIdx.x * 8 + wid;
    if (row >= nrows) return;
    float4* p = ((float4*)emb) + (size_t)row * (DIM / 4) + lane;
    float4 v = *p;
    float s = v.x * v.x + v.y * v.y + v.z * v.z + v.w * v.w;
    #pragma unroll
    for (int m = 16; m >= 1; m >>= 1) s += __shfl_xor(s, m, 32);
    const float inv = 1.0f / fmaxf(sqrtf(s), 1e-12f);
    v.x *= inv; v.y *= inv; v.z *= inv; v.w *= inv;
    *p = v;
}

// ---------------------------------------------------------------------------
extern "C" void kernel_launch(void* const* d_in, const int* in_sizes, int n_in,
                              void* d_out, int out_size, void* d_ws, size_t ws_size,
                              hipStream_t stream)
{
    const float* emb_in  = (const float*)d_in[0];   // [N*128] f32
    const int*   u_idx   = (const int*)d_in[1];     // [T*B]   i32
    const int*   tgt_idx = (const int*)d_in[2];     // [T*B*K] i32
    float* emb = (float*)d_out;                     // working table == output
    float* err = (float*)d_ws;                      // [BATCH*DIM] scratch

    const int N = in_sizes[0] / DIM;

    // working copy of the embedding table (graph-capture safe async d2d)
    hipMemcpyAsync(emb, emb_in, (size_t)N * DIM * sizeof(float),
                   hipMemcpyDeviceToDevice, stream);

    // 100 serial SGD steps: compute errors (WMMA), then atomic scatter
    for (int t = 0; t < NSTEPS; ++t) {
        line_compute_err<<<BATCH / 16, 256, 0, stream>>>(
            emb, u_idx + (size_t)t * BATCH,
            tgt_idx + (size_t)t * BATCH * KTGT, err);
        line_scatter<<<(BATCH * DIM) / 256, 256, 0, stream>>>(
            emb, u_idx + (size_t)t * BATCH, err);
    }

    // final L2 row-normalization (bandwidth bound: ~1 GB @ 23.3 TB/s)
    line_normalize<<<(N + 7) / 8, 256, 0, stream>>>(emb, N);
}